// KNNGaussianBlur_19207093748448
// MI455X (gfx1250) — compile-verified
//
#include <hip/hip_runtime.h>

// Fused separable Gaussian blur (sigma=4, 25 taps, replicate padding) for
// [3,4096,4096] fp32, using V_WMMA_F32_16X16X4_F32 with Toeplitz weight
// matrices. blur(img/max)*max == blur(img) by linearity (weights sum to 1),
// so the global-max pass is elided.

#define IMG_H 4096
#define IMG_W 4096
#define TH    64            // output tile rows per workgroup
#define TW    64            // output tile cols per workgroup
#define HALF  12            // 25-tap kernel radius
#define XROWS (TH + 2*HALF) // 88 input rows in LDS
#define XCOLS 96            // 6 col-blocks of 16 (>= TW + 2*HALF = 88)
#define XP    104           // LDS row pitch (even -> 8B/16B-aligned rows)
#define VP    104

typedef float v2f __attribute__((ext_vector_type(2)));
typedef float v8f __attribute__((ext_vector_type(8)));

// Types matching the async-to-LDS builtin signature revealed by the compiler:
// param0 = int __vector(4) __device__* (global), param1 = LDS side.
typedef int v4i_mem __attribute__((vector_size(16)));
typedef __attribute__((address_space(1))) v4i_mem* gv4i_p;
typedef __attribute__((address_space(3))) v4i_mem* lv4i_p;

// Normalized 25-tap Gaussian (sigma=4), w[t] = exp(-((t-12)/4)^2/2) / sum,
// padded with zeros to 32 entries so out-of-band Toeplitz indices read 0.
__device__ __constant__ float GW32[32] = {
    0.00110988f, 0.00227733f, 0.00438966f, 0.00794866f, 0.01352113f,
    0.02160670f, 0.03243550f, 0.04574138f, 0.06059749f, 0.07541479f,
    0.08816881f, 0.09683453f, 0.09990836f, 0.09683453f, 0.08816881f,
    0.07541479f, 0.06059749f, 0.04574138f, 0.03243550f, 0.02160670f,
    0.01352113f, 0.00794866f, 0.00438966f, 0.00227733f, 0.00110988f,
    0.0f, 0.0f, 0.0f, 0.0f, 0.0f, 0.0f, 0.0f };

__device__ __forceinline__ float wsel(int t) {
    unsigned u = (unsigned)t;
    float v = GW32[u & 31u];            // always in-bounds
    return (u < 32u) ? v : 0.0f;        // negative t / t>=32 -> 0
}

#if defined(__has_builtin)
#  if __has_builtin(__builtin_amdgcn_global_load_async_to_lds_b128)
#    define HAVE_ASYNC_LDS 1
#  endif
#endif
#ifndef HAVE_ASYNC_LDS
#  define HAVE_ASYNC_LDS 0
#endif

__device__ __forceinline__ void wait_async_zero() {
#if defined(__has_builtin) && __has_builtin(__builtin_amdgcn_s_wait_asynccnt)
    __builtin_amdgcn_s_wait_asynccnt(0);
#else
    asm volatile("s_wait_asynccnt 0" ::: "memory");
#endif
}

__global__ void __launch_bounds__(256)
gauss_blur_fused_wmma(const float* __restrict__ img, float* __restrict__ out) {
    __shared__ float Xs[XROWS * XP];   // input halo tile
    __shared__ float Vs[TH * VP];      // vertically blurred intermediate

    const int tid  = threadIdx.x;
    const int lane = tid & 31;
    const int wave = tid >> 5;
    const int ch   = blockIdx.z;
    const int r0   = blockIdx.y * TH;
    const int c0   = blockIdx.x * TW;

    const float* __restrict__ src = img + (size_t)ch * IMG_H * IMG_W;
    float* __restrict__ dst       = out + (size_t)ch * IMG_H * IMG_W;

    // ---- fill X tile (issue loads first so they overlap weight setup) ----
    const bool interior = (blockIdx.x > 0) & (blockIdx.x < gridDim.x - 1) &
                          (blockIdx.y > 0) & (blockIdx.y < gridDim.y - 1);
    bool used_async = false;
    if (interior) {
        // fully in-bounds; 16B-aligned (c0-12 is a multiple of 4)
        const float* base = src + (size_t)(r0 - HALF) * IMG_W + (c0 - HALF);
        for (int idx = tid; idx < XROWS * (XCOLS / 4); idx += 256) {
            int r  = idx / (XCOLS / 4);
            int c4 = idx - r * (XCOLS / 4);
            const float* gp = base + (size_t)r * IMG_W + 4 * c4;
            float*       lp = &Xs[r * XP + 4 * c4];
#if HAVE_ASYNC_LDS
            __builtin_amdgcn_global_load_async_to_lds_b128(
                (gv4i_p)gp, (lv4i_p)lp, 0, 0);
#else
            *(float4*)lp = *(const float4*)gp;
#endif
        }
#if HAVE_ASYNC_LDS
        used_async = true;
#endif
    } else {
        // edge tiles: scalar loads with replicate (clamp-to-edge) padding
        for (int idx = tid; idx < XROWS * XCOLS; idx += 256) {
            int r  = idx / XCOLS;
            int c  = idx - r * XCOLS;
            int gr = min(max(r0 - HALF + r, 0), IMG_H - 1);
            int gc = min(max(c0 - HALF + c, 0), IMG_W - 1);
            Xs[r * XP + c] = src[(size_t)gr * IMG_W + gc];
        }
    }

    // ---- per-lane Toeplitz weight registers (shared by both passes) ----
    // A-operand layout (16x4 f32): lane -> (m = lane&15, k = 2*(lane>>4)+{0,1})
    // B-operand layout (4x16 f32): lane -> (n = lane&15, k = 2*(lane>>4)+{0,1})
    const int mn = lane & 15;
    const int kb = (lane >> 4) * 2;
    v2f wt[10];
    #pragma unroll
    for (int q = 0; q < 10; ++q) {
        int t0 = 4 * q + kb - mn;
        wt[q].x = wsel(t0);
        wt[q].y = wsel(t0 + 1);
    }

    if (used_async) {
        wait_async_zero();
    }
    __syncthreads();

    // ---- vertical pass: V[m][u] = sum_t w[t] * X[m+t][u] ----
    // D(16x16) += A_q(16x4 Toeplitz weights) x B_q(4x16 input rows), q=0..9
    // 24 blocks (4 row-blocks x 6 col-blocks), 3 per wave (wave-uniform).
    #pragma unroll
    for (int bi = 0; bi < 3; ++bi) {
        int b  = wave * 3 + bi;
        int m0 = (b / 6) * 16;
        int u0 = (b % 6) * 16;
        v8f acc = {};
        #pragma unroll
        for (int q = 0; q < 10; ++q) {
            int j = m0 + 4 * q + kb;
            v2f bd;
            bd.x = Xs[(j    ) * XP + u0 + mn];
            bd.y = Xs[(j + 1) * XP + u0 + mn];
            acc = __builtin_amdgcn_wmma_f32_16x16x4_f32(
                false, wt[q], false, bd, (short)0, acc, false, false);
        }
        // D layout: VGPR i -> row m0 + i + 8*(lane>>4), col u0 + (lane&15)
        int vr = m0 + (lane >> 4) * 8;
        #pragma unroll
        for (int i = 0; i < 8; ++i)
            Vs[(vr + i) * VP + u0 + mn] = acc[i];
    }

    __syncthreads();

    // ---- horizontal pass: out[m][n] = sum_t w[t] * V[m][n+t] ----
    // D(16x16) += A_q(16x4 of V data) x B_q(4x16 Toeplitz weights), q=0..9
    // 16 blocks (4x4), 2 per wave.
    #pragma unroll
    for (int bi = 0; bi < 2; ++bi) {
        int b  = wave * 2 + bi;
        int m0 = (b / 4) * 16;
        int n0 = (b % 4) * 16;
        v8f acc = {};
        const float* vrow = &Vs[(m0 + mn) * VP + n0];
        #pragma unroll
        for (int q = 0; q < 10; ++q) {
            int u = 4 * q + kb;                 // even -> 8B aligned
            v2f ad = *(const v2f*)(vrow + u);   // ds_load_b64
            acc = __builtin_amdgcn_wmma_f32_16x16x4_f32(
                false, ad, false, wt[q], (short)0, acc, false, false);
        }
        int orow = r0 + m0 + (lane >> 4) * 8;
        int ocol = c0 + n0 + mn;
        #pragma unroll
        for (int i = 0; i < 8; ++i)
            dst[(size_t)(orow + i) * IMG_W + ocol] = acc[i];
    }
}

extern "C" void kernel_launch(void* const* d_in, const int* in_sizes, int n_in,
                              void* d_out, int out_size, void* d_ws, size_t ws_size,
                              hipStream_t stream) {
    (void)in_sizes; (void)n_in; (void)out_size; (void)d_ws; (void)ws_size;
    const float* img = (const float*)d_in[0];   // [1,3,4096,4096] fp32
    float* out       = (float*)d_out;           // [3,4096,4096] fp32
    dim3 grid(IMG_W / TW, IMG_H / TH, 3);       // 64 x 64 x 3 workgroups
    gauss_blur_fused_wmma<<<grid, 256, 0, stream>>>(img, out);
}